// MultimodalGNN_53601191854315
// MI455X (gfx1250) — compile-verified
//
#include <hip/hip_runtime.h>
#include <hip/hip_bf16.h>

typedef __attribute__((ext_vector_type(2))) float v2f;
typedef __attribute__((ext_vector_type(8))) float v8f;

#define NN 100000
#define EE 1200000
#define GG 256
#define DD 64

// ---------------- degree / norm prep ----------------

__global__ void k_init_deg(float* __restrict__ deg, int n) {
    int i = blockIdx.x * blockDim.x + threadIdx.x;
    if (i < n) deg[i] = 1.0f;   // self-loop contributes 1 to every node's degree
}

__global__ void k_count_deg(const int* __restrict__ dst, float* __restrict__ deg, int e) {
    int i = blockIdx.x * blockDim.x + threadIdx.x;
    if (i < e) atomicAdd(&deg[dst[i]], 1.0f);
}

__global__ void k_dinv(const float* __restrict__ deg, float* __restrict__ dinv, int n) {
    int i = blockIdx.x * blockDim.x + threadIdx.x;
    if (i < n) {
        float d = deg[i];
        dinv[i] = (d > 0.0f) ? rsqrtf(d) : 0.0f;
    }
}

__global__ void k_norm(const int* __restrict__ src, const int* __restrict__ dst,
                       const float* __restrict__ dinv, float* __restrict__ norm, int e) {
    int i = blockIdx.x * blockDim.x + threadIdx.x;
    if (i < e) norm[i] = dinv[src[i]] * dinv[dst[i]];
}

// ---------------- GEMM: O[n,64] = H[n,64] @ W[64,64] via FP32 WMMA ----------------
// One wave per 16-row tile; W staged in LDS; 16 x (16x16x4) WMMA per 16-col tile.

__global__ __launch_bounds__(256) void k_gemm64_wmma(const float* __restrict__ H,
                                                     const float* __restrict__ W,
                                                     float* __restrict__ O, int nrows) {
    __shared__ float Ws[DD * DD];
    for (int i = threadIdx.x; i < DD * DD; i += 256) Ws[i] = W[i];
    __syncthreads();

    const int wave = threadIdx.x >> 5;            // 0..7
    const int lane = threadIdx.x & 31;
    const int tile = blockIdx.x * 8 + wave;       // 16-row tile id (wave-uniform)
    const int row0 = tile * 16;
    if (row0 >= nrows) return;                    // wave-uniform, EXEC stays all-1 for WMMA

    const int lr = lane & 15;                     // row (A) / col (B,C)
    const int lh = lane >> 4;                     // lane-half selects K pair / M half
    const float* __restrict__ arow = H + (size_t)(row0 + lr) * DD;

    #pragma unroll
    for (int nt = 0; nt < 4; ++nt) {              // 4 column tiles of 16 -> D=64
        const int col = nt * 16 + lr;
        v8f acc = {};
        #pragma unroll
        for (int kc = 0; kc < 16; ++kc) {         // K = 64 = 16 * 4
            const int k0 = kc * 4 + lh * 2;
            v2f a, b;
            a.x = arow[k0];
            a.y = arow[k0 + 1];
            b.x = Ws[k0 * DD + col];
            b.y = Ws[(k0 + 1) * DD + col];
            acc = __builtin_amdgcn_wmma_f32_16x16x4_f32(
                false, a, false, b, (short)0, acc, false, false);
        }
        // C/D layout: VGPR r, lanes 0-15 -> M=r, lanes 16-31 -> M=r+8, N=lane%16
        #pragma unroll
        for (int r = 0; r < 8; ++r) {
            O[(size_t)(row0 + r + lh * 8) * DD + col] = acc[r];
        }
    }
}

// ---------------- aggregation ----------------

// agg[i,:] = hw[i,:] * dinv[i]^2   (self-loop term; also serves as the init write)
__global__ void k_self_init(const float* __restrict__ hw, const float* __restrict__ dinv,
                            float* __restrict__ agg, int n) {
    size_t idx = (size_t)blockIdx.x * blockDim.x + threadIdx.x;
    if (idx < (size_t)n * DD) {
        int i = (int)(idx >> 6);
        float di = dinv[i];
        agg[idx] = hw[idx] * di * di;
    }
}

// agg[dst,:] += hw[src,:] * norm[e]   one thread per (edge, feature)
__global__ void k_scatter_edges(const int* __restrict__ src, const int* __restrict__ dst,
                                const float* __restrict__ norm, const float* __restrict__ hw,
                                float* __restrict__ agg, int e) {
    size_t idx = (size_t)blockIdx.x * blockDim.x + threadIdx.x;
    if (idx < (size_t)e * DD) {
        int ei = (int)(idx >> 6);
        int d  = (int)(idx & 63);
        int s  = src[ei];
        int t  = dst[ei];
        float v = hw[(size_t)s * DD + d] * norm[ei];
        atomicAdd(&agg[(size_t)t * DD + d], v);
    }
}

__global__ void k_bias_relu(float* __restrict__ h, const float* __restrict__ b, int n) {
    size_t idx = (size_t)blockIdx.x * blockDim.x + threadIdx.x;
    if (idx < (size_t)n * DD) {
        float v = h[idx] + b[idx & 63];
        h[idx] = fmaxf(v, 0.0f);
    }
}

// ---------------- pooling ----------------

__global__ void k_pool_init(float* __restrict__ sums, float* __restrict__ cnts) {
    int i = blockIdx.x * blockDim.x + threadIdx.x;
    if (i < GG * DD) sums[i] = 0.0f;
    if (i < GG) cnts[i] = 0.0f;
}

__global__ void k_pool_scatter(const float* __restrict__ h, const int* __restrict__ batch,
                               float* __restrict__ sums, float* __restrict__ cnts, int n) {
    size_t idx = (size_t)blockIdx.x * blockDim.x + threadIdx.x;
    if (idx < (size_t)n * DD) {
        int i = (int)(idx >> 6);
        int d = (int)(idx & 63);
        int g = batch[i];
        atomicAdd(&sums[(size_t)g * DD + d], h[idx]);
        if (d == 0) atomicAdd(&cnts[g], 1.0f);
    }
}

// ---------------- heads: one thread per graph ----------------

__global__ __launch_bounds__(256) void k_heads(const float* __restrict__ sums,
                                               const float* __restrict__ cnts,
                                               const float* __restrict__ Wc1, const float* __restrict__ bc1,
                                               const float* __restrict__ Wc2, const float* __restrict__ bc2,
                                               const float* __restrict__ Wm1, const float* __restrict__ bm1,
                                               const float* __restrict__ Wm2, const float* __restrict__ bm2,
                                               float* __restrict__ out) {
    int g = blockIdx.x * blockDim.x + threadIdx.x;
    if (g >= GG) return;
    float inv = 1.0f / fmaxf(cnts[g], 1.0f);
    float ge[DD];
    #pragma unroll
    for (int d = 0; d < DD; ++d) ge[d] = sums[(size_t)g * DD + d] * inv;

    float c[32];
    #pragma unroll
    for (int j = 0; j < 32; ++j) {
        float s = bc1[j];
        for (int d = 0; d < DD; ++d) s += ge[d] * Wc1[d * 32 + j];
        c[j] = fmaxf(s, 0.0f);
    }
    #pragma unroll
    for (int k = 0; k < 2; ++k) {
        float s = bc2[k];
        for (int j = 0; j < 32; ++j) s += c[j] * Wc2[j * 2 + k];
        out[g * 2 + k] = s;                 // logits: out[0 .. 2G)
    }

    float m[32];
    #pragma unroll
    for (int j = 0; j < 32; ++j) {
        float s = bm1[j];
        for (int d = 0; d < DD; ++d) s += ge[d] * Wm1[d * 32 + j];
        m[j] = fmaxf(s, 0.0f);
    }
    float s = bm2[0];
    for (int j = 0; j < 32; ++j) s += m[j] * Wm2[j];
    out[2 * GG + g] = s;                    // mag: out[2G .. 3G)
}

// ---------------- launch ----------------

extern "C" void kernel_launch(void* const* d_in, const int* in_sizes, int n_in,
                              void* d_out, int out_size, void* d_ws, size_t ws_size,
                              hipStream_t stream) {
    const float* x   = (const float*)d_in[0];
    const int*   ei  = (const int*)d_in[1];        // [2, E] flat: src then dst
    const int*   bat = (const int*)d_in[2];
    const float* W1  = (const float*)d_in[3];
    const float* b1  = (const float*)d_in[4];
    const float* W2  = (const float*)d_in[5];
    const float* b2  = (const float*)d_in[6];
    const float* Wc1 = (const float*)d_in[7];
    const float* bc1 = (const float*)d_in[8];
    const float* Wc2 = (const float*)d_in[9];
    const float* bc2 = (const float*)d_in[10];
    const float* Wm1 = (const float*)d_in[11];
    const float* bm1 = (const float*)d_in[12];
    const float* Wm2 = (const float*)d_in[13];
    const float* bm2 = (const float*)d_in[14];
    float* out = (float*)d_out;

    const int* src = ei;
    const int* dst = ei + EE;

    // workspace carve-out (256B aligned blocks)
    char* ws = (char*)d_ws;
    size_t off = 0;
    auto carve = [&](size_t bytes) -> char* {
        char* p = ws + off;
        off += (bytes + 255) & ~(size_t)255;
        return p;
    };
    float* deg  = (float*)carve((size_t)NN * 4);
    float* dinv = (float*)carve((size_t)NN * 4);
    float* nrm  = (float*)carve((size_t)EE * 4);
    float* bufA = (float*)carve((size_t)NN * DD * 4);   // hw (GEMM out)
    float* bufB = (float*)carve((size_t)NN * DD * 4);   // agg / h
    float* sums = (float*)carve((size_t)GG * DD * 4);
    float* cnts = (float*)carve((size_t)GG * 4);
    (void)ws_size;

    const int T = 256;
    const int gN   = (NN + T - 1) / T;
    const int gE   = (EE + T - 1) / T;
    const int gND  = (int)(((size_t)NN * DD + T - 1) / T);
    const int gED  = (int)(((size_t)EE * DD + T - 1) / T);
    const int gGmm = ((NN / 16) + 7) / 8;               // 6250 tiles / 8 waves

    // degrees + normalization
    k_init_deg<<<gN, T, 0, stream>>>(deg, NN);
    k_count_deg<<<gE, T, 0, stream>>>(dst, deg, EE);
    k_dinv<<<gN, T, 0, stream>>>(deg, dinv, NN);
    k_norm<<<gE, T, 0, stream>>>(src, dst, dinv, nrm, EE);

    // layer 1: bufA = x @ W1 ; bufB = A_hat_norm @ bufA ; relu(+b1)
    k_gemm64_wmma<<<gGmm, T, 0, stream>>>(x, W1, bufA, NN);
    k_self_init<<<gND, T, 0, stream>>>(bufA, dinv, bufB, NN);
    k_scatter_edges<<<gED, T, 0, stream>>>(src, dst, nrm, bufA, bufB, EE);
    k_bias_relu<<<gND, T, 0, stream>>>(bufB, b1, NN);

    // layer 2: bufA = bufB @ W2 ; bufB = A_hat_norm @ bufA ; relu(+b2)
    k_gemm64_wmma<<<gGmm, T, 0, stream>>>(bufB, W2, bufA, NN);
    k_self_init<<<gND, T, 0, stream>>>(bufA, dinv, bufB, NN);
    k_scatter_edges<<<gED, T, 0, stream>>>(src, dst, nrm, bufA, bufB, EE);
    k_bias_relu<<<gND, T, 0, stream>>>(bufB, b2, NN);

    // mean pool + heads
    k_pool_init<<<(GG * DD + T - 1) / T, T, 0, stream>>>(sums, cnts);
    k_pool_scatter<<<gND, T, 0, stream>>>(bufB, bat, sums, cnts, NN);
    k_heads<<<1, T, 0, stream>>>(sums, cnts, Wc1, bc1, Wc2, bc2, Wm1, bm1, Wm2, bm2, out);
    (void)out_size; (void)n_in; (void)in_sizes;
}